// DCRNNEncoder_10290741641297
// MI455X (gfx1250) — compile-verified
//
#include <hip/hip_runtime.h>
#include <math.h>

typedef __attribute__((ext_vector_type(2))) float v2f;
typedef __attribute__((ext_vector_type(8))) float v8f;

// Problem constants
#define NNODES 512
#define BATCH  32
#define HID    64
#define TSTEPS 48
// plane = (NNODES, BATCH, HID) node-major activations
#define PLANE_ELEMS ((size_t)NNODES * BATCH * HID)   // 1,048,576

// ---------------------------------------------------------------------------
// Layout transforms: (b, n, c) <-> (n, b, c), vectorized by float4 over c.
// ---------------------------------------------------------------------------
__global__ __launch_bounds__(256) void nbc_from_bnc(float4* __restrict__ dst,
                                                    const float4* __restrict__ src) {
    int tid = blockIdx.x * 256 + threadIdx.x;           // 0 .. 262143
    int c4 = tid & 15, b = (tid >> 4) & 31, n = tid >> 9;
    dst[((n << 5) + b) * 16 + c4] = src[((b << 9) + n) * 16 + c4];
}
__global__ __launch_bounds__(256) void bnc_from_nbc(float4* __restrict__ dst,
                                                    const float4* __restrict__ src) {
    int tid = blockIdx.x * 256 + threadIdx.x;
    int c4 = tid & 15, b = (tid >> 4) & 31, n = tid >> 9;
    dst[((b << 9) + n) * 16 + c4] = src[((n << 5) + b) * 16 + c4];
}

// ---------------------------------------------------------------------------
// Diffusion GEMM: C(512 x 2048) = S(512x512) @ P(512x2048)
// cheb != 0: C = 2*(S@P) - Q    (Chebyshev recurrence x_k = 2 S x_{k-1} - x_{k-2})
// Block = 256 threads (8 waves), tile 64(M) x 128(N), K staged 32 at a time.
// Waves in 2(M) x 4(N) grid; each wave owns a 32x32 area = 2x2 WMMA tiles so
// every A/B fragment is shared by two v_wmma_f32_16x16x4_f32 ops.
// ---------------------------------------------------------------------------
__global__ __launch_bounds__(256) void gemm_s(
    float* __restrict__ C, const float* __restrict__ S,
    const float* __restrict__ P, const float* __restrict__ Q, int cheb)
{
    __shared__ float As[64][34];    // padded: fragment rows hit distinct banks
    __shared__ float Bs[32][136];   // pad so half-wave K-offset rows don't collide

    const int tid  = threadIdx.x;
    const int nb   = blockIdx.x;    // 0..15  (128-col block)
    const int mb   = blockIdx.y;    // 0..7   (64-row block)
    const int wid  = tid >> 5, lane = tid & 31;
    const int half = lane >> 4, idx = lane & 15;
    const int rowW = (wid & 1) * 32;    // wave M offset within block
    const int colW = (wid >> 1) * 32;   // wave N offset within block

    v8f acc[2][2] = {};

    for (int k0 = 0; k0 < 512; k0 += 32) {
#pragma unroll
        for (int i = 0; i < 2; i++) {   // A tile: 64x32, two float4 per thread
            int f = tid + i * 256;
            int r = f >> 3, c4 = (f & 7) * 4;
            float4 v = *(const float4*)&S[(size_t)(mb * 64 + r) * 512 + k0 + c4];
            As[r][c4 + 0] = v.x; As[r][c4 + 1] = v.y;
            As[r][c4 + 2] = v.z; As[r][c4 + 3] = v.w;
        }
#pragma unroll
        for (int i = 0; i < 4; i++) {   // B tile: 32x128, four float4 per thread
            int f = tid + i * 256;
            int r = f >> 5, c4 = (f & 31) * 4;
            float4 v = *(const float4*)&P[(size_t)(k0 + r) * 2048 + nb * 128 + c4];
            Bs[r][c4 + 0] = v.x; Bs[r][c4 + 1] = v.y;
            Bs[r][c4 + 2] = v.z; Bs[r][c4 + 3] = v.w;
        }
        __syncthreads();
#pragma unroll
        for (int kk = 0; kk < 32; kk += 4) {
            int ka = kk + 2 * half;     // per-lane K base for f32 16x16x4 fragments
            v2f a0; a0.x = As[rowW + idx][ka];      a0.y = As[rowW + idx][ka + 1];
            v2f a1; a1.x = As[rowW + 16 + idx][ka]; a1.y = As[rowW + 16 + idx][ka + 1];
            v2f b0; b0.x = Bs[ka][colW + idx];      b0.y = Bs[ka + 1][colW + idx];
            v2f b1; b1.x = Bs[ka][colW + 16 + idx]; b1.y = Bs[ka + 1][colW + 16 + idx];
            acc[0][0] = __builtin_amdgcn_wmma_f32_16x16x4_f32(false, a0, false, b0,
                                                              (short)0, acc[0][0], false, false);
            acc[0][1] = __builtin_amdgcn_wmma_f32_16x16x4_f32(false, a0, false, b1,
                                                              (short)0, acc[0][1], false, false);
            acc[1][0] = __builtin_amdgcn_wmma_f32_16x16x4_f32(false, a1, false, b0,
                                                              (short)0, acc[1][0], false, false);
            acc[1][1] = __builtin_amdgcn_wmma_f32_16x16x4_f32(false, a1, false, b1,
                                                              (short)0, acc[1][1], false, false);
        }
        __syncthreads();
    }
#pragma unroll
    for (int mt = 0; mt < 2; mt++) {
#pragma unroll
        for (int v = 0; v < 8; v++) {
            int r = mb * 64 + rowW + mt * 16 + v + 8 * half;
            size_t o0 = (size_t)r * 2048 + nb * 128 + colW + idx;
            size_t o1 = o0 + 16;
            float x0 = acc[mt][0][v], x1 = acc[mt][1][v];
            if (cheb) { x0 = 2.f * x0 - Q[o0]; x1 = 2.f * x1 - Q[o1]; }
            C[o0] = x0; C[o1] = x1;
        }
    }
}

// ---------------------------------------------------------------------------
// Projection GEMM: Out(16384 x OD) = [P0..P5](16384 x 384) @ W(384 x OD) + bias
// Feature planes are the K blocks (64 each) in reference concat order.
// Block = 256 threads (8 waves), tile 256(M) x OD(N); each wave owns 32 rows
// (2 M-tiles) x OD cols: each A fragment feeds OD/16 WMMAs, each B fragment 2.
// ---------------------------------------------------------------------------
template <int OD>
__global__ __launch_bounds__(256) void gemm_w(
    float* __restrict__ Out,
    const float* __restrict__ P0, const float* __restrict__ P1,
    const float* __restrict__ P2, const float* __restrict__ P3,
    const float* __restrict__ P4, const float* __restrict__ P5,
    const float* __restrict__ W, const float* __restrict__ bias)
{
    constexpr int BSTR = OD + 8;
    constexpr int NT   = OD / 16;
    __shared__ float As[256][34];
    __shared__ float Bs[32][BSTR];

    const float* planes[6] = {P0, P1, P2, P3, P4, P5};
    const int tid  = threadIdx.x;
    const int rb   = blockIdx.x;            // 0..63 row blocks of 256
    const int wid  = tid >> 5, lane = tid & 31;
    const int half = lane >> 4, idx = lane & 15;
    const int rowW = wid * 32;

    v8f acc[2][NT] = {};

    for (int k0 = 0; k0 < 384; k0 += 32) {
        const float* pl = planes[k0 >> 6];
        const int colb  = k0 & 63;
#pragma unroll
        for (int i = 0; i < 8; i++) {       // A tile: 256x32
            int f = tid + i * 256;
            int r = f >> 3, c4 = (f & 7) * 4;
            float4 v = *(const float4*)&pl[(size_t)(rb * 256 + r) * 64 + colb + c4];
            As[r][c4 + 0] = v.x; As[r][c4 + 1] = v.y;
            As[r][c4 + 2] = v.z; As[r][c4 + 3] = v.w;
        }
        constexpr int BF4 = (32 * OD) / 4;
#pragma unroll
        for (int i = 0; i < BF4 / 256; i++) {   // B tile: 32 x OD from W
            int f = tid + i * 256;
            int r = f / (OD / 4), c4 = (f % (OD / 4)) * 4;
            float4 v = *(const float4*)&W[(size_t)(k0 + r) * OD + c4];
            Bs[r][c4 + 0] = v.x; Bs[r][c4 + 1] = v.y;
            Bs[r][c4 + 2] = v.z; Bs[r][c4 + 3] = v.w;
        }
        __syncthreads();
#pragma unroll
        for (int kk = 0; kk < 32; kk += 4) {
            int ka = kk + 2 * half;
            v2f a0; a0.x = As[rowW + idx][ka];      a0.y = As[rowW + idx][ka + 1];
            v2f a1; a1.x = As[rowW + 16 + idx][ka]; a1.y = As[rowW + 16 + idx][ka + 1];
#pragma unroll
            for (int t = 0; t < NT; t++) {
                v2f b; b.x = Bs[ka][t * 16 + idx]; b.y = Bs[ka + 1][t * 16 + idx];
                acc[0][t] = __builtin_amdgcn_wmma_f32_16x16x4_f32(false, a0, false, b,
                                                                  (short)0, acc[0][t], false, false);
                acc[1][t] = __builtin_amdgcn_wmma_f32_16x16x4_f32(false, a1, false, b,
                                                                  (short)0, acc[1][t], false, false);
            }
        }
        __syncthreads();
    }
#pragma unroll
    for (int t = 0; t < NT; t++) {
        float bb = bias[t * 16 + idx];
#pragma unroll
        for (int mt = 0; mt < 2; mt++) {
#pragma unroll
            for (int v = 0; v < 8; v++) {
                int r = rb * 256 + rowW + mt * 16 + v + 8 * half;
                Out[(size_t)r * OD + t * 16 + idx] = acc[mt][t][v] + bb;
            }
        }
    }
}

// ---------------------------------------------------------------------------
// GRU pointwise kernels
// ---------------------------------------------------------------------------
__global__ __launch_bounds__(256) void gate_elem(
    float* __restrict__ RH, float* __restrict__ U,
    const float* __restrict__ GA, const float* __restrict__ Hp)
{
    int tid = blockIdx.x * 256 + threadIdx.x;   // 0 .. PLANE_ELEMS-1
    int r = tid >> 6, c = tid & 63;
    float g1 = GA[(size_t)r * 128 + c];
    float g2 = GA[(size_t)r * 128 + 64 + c];
    float rs = 1.f / (1.f + expf(-g1));
    float us = 1.f / (1.f + expf(-g2));
    RH[tid] = rs * Hp[tid];
    U[tid]  = us;
}

__global__ __launch_bounds__(256) void newh_elem(
    float* __restrict__ Hp, float* __restrict__ seq_t,
    const float* __restrict__ Cc, const float* __restrict__ U)
{
    int tid = blockIdx.x * 256 + threadIdx.x;
    float cv = tanhf(Cc[tid]);
    float u  = U[tid], h = Hp[tid];
    float nh = u * h + (1.f - u) * cv;
    Hp[tid] = nh;
    int c = tid & 63, r = tid >> 6, b = r & 31, n = r >> 5;
    seq_t[((size_t)b << 15) + ((size_t)n << 6) + c] = nh;  // (b, n*64 + c)
}

// ---------------------------------------------------------------------------
// Host orchestration (all launches on `stream`; graph-capture safe)
// ---------------------------------------------------------------------------
extern "C" void kernel_launch(void* const* d_in, const int* in_sizes, int n_in,
                              void* d_out, int out_size, void* d_ws, size_t ws_size,
                              hipStream_t stream)
{
    (void)in_sizes; (void)n_in; (void)out_size; (void)ws_size;

    const float* inputs = (const float*)d_in[0];   // (48, 32, 512*64)
    const float* h0     = (const float*)d_in[1];   // (2, 32, 512*64)
    const float* S      = (const float*)d_in[2];   // (1, 512, 512)
    const float* Wg     = (const float*)d_in[3];   // (2, 384, 128)
    const float* bg     = (const float*)d_in[4];   // (2, 128)
    const float* Wc     = (const float*)d_in[5];   // (2, 384, 64)
    const float* bc     = (const float*)d_in[6];   // (2, 64)
    float* out = (float*)d_out;

    const size_t PL = PLANE_ELEMS;
    float* ws  = (float*)d_ws;
    float* Xp  = ws;            // x plane           (n,b,64)
    float* Hp  = ws + 1 * PL;   // h plane
    float* SX  = ws + 2 * PL;   // S @ x
    float* SH  = ws + 3 * PL;   // S @ h   / S @ rh
    float* T2X = ws + 4 * PL;   // 2 S Sx - x
    float* T2H = ws + 5 * PL;   // 2 S Sh - h / 2 S S rh - rh
    float* RH  = ws + 6 * PL;   // r * h
    float* U   = ws + 7 * PL;   // update gate
    float* GA  = ws + 8 * PL;   // gate preact (16384 x 128)  [2 planes]
    float* Cc  = ws + 10 * PL;  // candidate preact (16384 x 64)

    float* finalH = out;                                    // (2, 32, 512*64)
    float* seq    = out + 2 * (size_t)BATCH * NNODES * HID; // (48, 32, 512*64)

    const dim3 bs(256);
    const dim3 g_nbc((unsigned)(PL / 4 / 256));   // 1024
    const dim3 g_el((unsigned)(PL / 256));        // 4096
    const dim3 g_gs(16, 8);                       // gemm_s: 2048/128 x 512/64
    const dim3 g_gw(64);                          // gemm_w: 16384/256

    for (int l = 0; l < 2; l++) {
        const float* Wg_l = Wg + (size_t)l * 384 * 128;
        const float* bg_l = bg + (size_t)l * 128;
        const float* Wc_l = Wc + (size_t)l * 384 * 64;
        const float* bc_l = bc + (size_t)l * 64;

        nbc_from_bnc<<<g_nbc, bs, 0, stream>>>((float4*)Hp,
            (const float4*)(h0 + (size_t)l * BATCH * NNODES * HID));

        for (int t = 0; t < TSTEPS; t++) {
            const float* xin = (l == 0)
                ? inputs + (size_t)t * BATCH * NNODES * HID
                : seq    + (size_t)t * BATCH * NNODES * HID;

            nbc_from_bnc<<<g_nbc, bs, 0, stream>>>((float4*)Xp, (const float4*)xin);

            // ---- gconv #1 (gates); x-diffusion shared with gconv #2 ----
            gemm_s<<<g_gs, bs, 0, stream>>>(SX,  S, Xp, nullptr, 0); // S@x
            gemm_s<<<g_gs, bs, 0, stream>>>(SH,  S, Hp, nullptr, 0); // S@h
            gemm_s<<<g_gs, bs, 0, stream>>>(T2X, S, SX, Xp, 1);      // 2 S Sx - x
            gemm_s<<<g_gs, bs, 0, stream>>>(T2H, S, SH, Hp, 1);      // 2 S Sh - h
            gemm_w<128><<<g_gw, bs, 0, stream>>>(GA, Xp, Hp, SX, SH, T2X, T2H,
                                                 Wg_l, bg_l);
            gate_elem<<<g_el, bs, 0, stream>>>(RH, U, GA, Hp);

            // ---- gconv #2 (candidate); only h-half recomputed ----
            gemm_s<<<g_gs, bs, 0, stream>>>(SH,  S, RH, nullptr, 0); // S@rh
            gemm_s<<<g_gs, bs, 0, stream>>>(T2H, S, SH, RH, 1);      // 2 S S rh - rh
            gemm_w<64><<<g_gw, bs, 0, stream>>>(Cc, Xp, RH, SX, SH, T2X, T2H,
                                                Wc_l, bc_l);

            newh_elem<<<g_el, bs, 0, stream>>>(Hp,
                seq + (size_t)t * BATCH * NNODES * HID, Cc, U);
        }

        bnc_from_nbc<<<g_nbc, bs, 0, stream>>>(
            (float4*)(finalH + (size_t)l * BATCH * NNODES * HID), (const float4*)Hp);
    }
}